// ConvCaps_82480551952822
// MI455X (gfx1250) — compile-verified
//
#include <hip/hip_runtime.h>
#include <hip/hip_bf16.h>
#include <math.h>

// ---------------- problem constants ----------------
#define NB    16
#define B_T   32
#define C_T   32
#define KC    3
#define H     4
#define HH    16
#define WIN   14
#define WD    6
#define WW    36
#define BKK   288      // B_T * K * K
#define CWW   1152     // C_T * WW
#define NITER 3
#define EPSF  1e-10f
#define HALF_LN2PI 0.9189385332046727f

// workspace element counts (floats)
#define NP   (NB*BKK*HH*WW)   // patches P[b][bkk][k*4+j][wpos]
#define NAP  (NB*BKK*WW)      // activation patches
#define NR   (NB*BKK*CWW)     // routing R
#define NMU  (NB*CWW*HH)      // mu / sigma
#define NAV  (NB*CWW)         // a

typedef __attribute__((ext_vector_type(2))) float v2f;
typedef __attribute__((ext_vector_type(8))) float v8f;

// ---------------------------------------------------------------------------
// prep: gather patches P[b,bkk,(k*4+j),wpos], AP[b,bkk,wpos]; init R = 1/32
//   wpos = wcol*6 + wrow ; row = 2*wrow+kx ; col = 2*wcol+ky
// ---------------------------------------------------------------------------
__global__ __launch_bounds__(256) void prep_kernel(
    const float* __restrict__ poses, const float* __restrict__ act,
    float* __restrict__ P, float* __restrict__ AP, float* __restrict__ R)
{
  int idx = blockIdx.x * 256 + threadIdx.x;
  if (idx < NR) R[idx] = 1.0f / 32.0f;
  if (idx < NP) {
    int wpos = idx % WW;  int t = idx / WW;
    int hh   = t % HH;    t /= HH;
    int bkk  = t % BKK;   int b = t / BKK;
    int bt = bkk / 9, k9 = bkk % 9, kx = k9 / 3, ky = k9 % 3;
    int wcol = wpos / WD, wrow = wpos % WD;
    int row = 2 * wrow + kx, col = 2 * wcol + ky;
    P[idx] = poses[((size_t)(b * (HH * B_T) + hh * B_T + bt) * WIN + row) * WIN + col];
  }
  if (idx < NAP) {
    int wpos = idx % WW;  int t = idx / WW;
    int bkk  = t % BKK;   int b = t / BKK;
    int bt = bkk / 9, k9 = bkk % 9, kx = k9 / 3, ky = k9 % 3;
    int wcol = wpos / WD, wrow = wpos % WD;
    int row = 2 * wrow + kx, col = 2 * wcol + ky;
    AP[idx] = act[((size_t)(b * B_T + bt) * WIN + row) * WIN + col];
  }
}

// ---------------------------------------------------------------------------
// stats: per (b, wpos-group of 4). 8 waves = 8 tiles of 4 channels.
// Pipeline over 144 stages (2 bkk each): wave 0 DMA-stages the shared
// P-patch tile + AP into LDS via global_load_async_to_lds_b128 (ASYNCcnt,
// double-buffered), all waves recompute the 16x16 V tile with
// V_WMMA_F32_16X16X4_F32 and accumulate S0/S1/S2.
// D-tile layout (wave32): VGPR v, lane l: M = v + (l>=16?8:0), N = l&15.
//   M = c_sub*4 + i ; N = wps*4 + j
// ---------------------------------------------------------------------------
__global__ __launch_bounds__(256) void stats_kernel(
    const float* __restrict__ W, const float* __restrict__ P,
    const float* __restrict__ AP, const float* __restrict__ R,
    const float* __restrict__ beta_v, const float* __restrict__ beta_a,
    const float* __restrict__ lambda_p,
    float* __restrict__ muOut, float* __restrict__ sgOut, float* __restrict__ aOut)
{
  // double buffer: per stage 2 bkk tiles of (16x4 P floats + 4 AP floats)
  __shared__ float buf[2][136];

  const int b     = blockIdx.x / 9;
  const int wpos0 = (blockIdx.x % 9) * 4;
  const int wave  = threadIdx.x >> 5;
  const int lane  = threadIdx.x & 31;
  const int half  = lane >> 4;
  const int lN    = lane & 15;
  const int wps   = lN >> 2;
  const int j     = lN & 3;
  const int wpos  = wpos0 + wps;
  const int c0    = wave * 4;
  // A operand rows: lanes 0-15 hold M=lane K=0,1 ; lanes 16-31 M=lane-16 K=2,3
  const int cA_row = c0 + (lN >> 2);
  const int i_row  = lN & 3;
  const int k0     = half * 2;
  // accumulator-side channel mapping
  const int cA   = c0 + half * 2;
  const int cB   = cA + 1;
  const int cwwA = cA * WW + wpos;
  const int cwwB = cwwA + WW;

  const float* Pb  = P  + (size_t)b * BKK * HH * WW;
  const float* Rb  = R  + (size_t)b * BKK * CWW;
  const float* APb = AP + (size_t)b * BKK * WW;

  // wave-0 DMA: stage `stage` (bkk = 2*stage, 2*stage+1) into buf[stage&1]
  auto issue = [&](int stage) {
    const int bkk0  = stage * 2;
    const int buf_i = stage & 1;
    {
      const int u = lane >> 4, hh = lane & 15;   // 32 lanes = 2 bkk x 16 rows
      const float* g = Pb + ((size_t)(bkk0 + u) * HH + hh) * WW + wpos0;
      unsigned lo = (unsigned)(uintptr_t)&buf[buf_i][u * 68 + hh * 4];
      asm volatile("global_load_async_to_lds_b128 %0, %1, off"
                   :: "v"(lo), "v"((unsigned long long)(uintptr_t)g) : "memory");
    }
    if (lane < 2) {                               // 2 lanes = 2 bkk AP rows
      const float* g = APb + (size_t)(bkk0 + lane) * WW + wpos0;
      unsigned lo = (unsigned)(uintptr_t)&buf[buf_i][lane * 68 + 64];
      asm volatile("global_load_async_to_lds_b128 %0, %1, off"
                   :: "v"(lo), "v"((unsigned long long)(uintptr_t)g) : "memory");
    }
  };

  if (wave == 0) { issue(0); issue(1); }

  v8f s1 = {}; v8f s2 = {};
  float s0A = 0.f, s0B = 0.f;

  for (int s = 0; s < BKK / 2; ++s) {
    if (wave == 0)  // stage s+1 outstanding (2 async ops) -> stage s landed
      asm volatile("s_wait_asynccnt 0x2" ::: "memory");
    __syncthreads();
    const float* lb = buf[s & 1];
    const int bkk0 = s * 2;
#pragma unroll
    for (int u = 0; u < 2; ++u) {
      const int bkk = bkk0 + u;
      const float* Wp = W + ((size_t)bkk * C_T + cA_row) * HH + i_row * 4 + k0;
      __builtin_prefetch(Wp + 2 * C_T * HH, 0, 1);
      v2f av; av.x = Wp[0]; av.y = Wp[1];
      // B operand from LDS: VGPR v, lanes0-15 K=v, lanes16-31 K=v+2
      v2f bv; bv.x = lb[u * 68 + (k0 * 4 + j) * 4 + wps];
              bv.y = lb[u * 68 + ((k0 + 1) * 4 + j) * 4 + wps];
      v8f cz = {};
      v8f d = __builtin_amdgcn_wmma_f32_16x16x4_f32(false, av, false, bv,
                                                    (short)0, cz, false, false);
      const float ai  = lb[u * 68 + 64 + wps];
      const float r4A = Rb[(size_t)bkk * CWW + cwwA] * ai;
      const float r4B = Rb[(size_t)bkk * CWW + cwwB] * ai;
      s0A += r4A; s0B += r4B;
#pragma unroll
      for (int v = 0; v < 8; ++v) {
        float r4 = (v < 4) ? r4A : r4B;
        float dv = d[v];
        s1[v] += r4 * dv;
        s2[v] += r4 * dv * dv;
      }
    }
    __syncthreads();                              // all done reading buf[s&1]
    if (wave == 0 && s + 2 < BKK / 2) issue(s + 2);
  }

  const float lam = lambda_p[0];
  const float bvA = beta_v[cA], bvB = beta_v[cB];
  float mu_[8], sg_[8];
  float costA = 0.f, costB = 0.f;
#pragma unroll
  for (int v = 0; v < 8; ++v) {
    float s0 = (v < 4) ? s0A : s0B;
    float m  = s1[v] / s0;
    float sq = s2[v] / s0 - m * m;
    sq = fmaxf(sq, 0.f);
    mu_[v] = m; sg_[v] = sq;
    float bvv = (v < 4) ? bvA : bvB;
    float cost = (bvv + logf(sqrtf(sq) + EPSF)) * s0;
    if (v < 4) costA += cost; else costB += cost;
  }

  const size_t baseA = ((size_t)b * CWW + cwwA) * HH + j;
  const size_t baseB = ((size_t)b * CWW + cwwB) * HH + j;
#pragma unroll
  for (int v = 0; v < 4; ++v) {
    muOut[baseA + v * 4] = mu_[v];      sgOut[baseA + v * 4] = sg_[v];
    muOut[baseB + v * 4] = mu_[v + 4];  sgOut[baseB + v * 4] = sg_[v + 4];
  }

  // reduce cost over j (4 consecutive lanes)
  costA += __shfl_xor(costA, 1, 4);  costA += __shfl_xor(costA, 2, 4);
  costB += __shfl_xor(costB, 1, 4);  costB += __shfl_xor(costB, 2, 4);
  if (j == 0) {
    float aA = 1.f / (1.f + expf(-lam * (beta_a[cA] - costA)));
    float aB = 1.f / (1.f + expf(-lam * (beta_a[cB] - costB)));
    aOut[(size_t)b * CWW + cwwA] = aA;
    aOut[(size_t)b * CWW + cwwB] = aB;
  }
}

// ---------------------------------------------------------------------------
// rupdate: per (b, bkk). Recompute V tiles via WMMA, Gaussian p, psum over HH,
// ap = a*psum; normalize over all 1152 cww via LDS reduction, R = ap/den + eps.
// ---------------------------------------------------------------------------
__global__ __launch_bounds__(256) void rupdate_kernel(
    const float* __restrict__ W, const float* __restrict__ P,
    const float* __restrict__ mu, const float* __restrict__ sg,
    const float* __restrict__ aV, float* __restrict__ R)
{
  __shared__ float red[256];
  const int b    = blockIdx.x / BKK;
  const int bkk  = blockIdx.x % BKK;
  const int wave = threadIdx.x >> 5;
  const int lane = threadIdx.x & 31;
  const int half = lane >> 4;
  const int lN   = lane & 15;
  const int wps  = lN >> 2;
  const int j    = lN & 3;
  const int c0   = wave * 4;
  const int cA   = c0 + half * 2;
  const int cB   = cA + 1;
  const int k0   = half * 2;

  const float* Wp = W + ((size_t)bkk * C_T + (c0 + (lN >> 2))) * HH + (lN & 3) * 4 + k0;
  v2f av; av.x = Wp[0]; av.y = Wp[1];                   // A constant over wpos loop
  const float* Pp = P + ((size_t)b * BKK + bkk) * HH * WW;
  float* Rrow = R + ((size_t)b * BKK + bkk) * CWW;
  const float* aVb = aV + (size_t)b * CWW;

  float den_loc = 0.f;
  for (int wg = 0; wg < 9; ++wg) {
    const int wpos = wg * 4 + wps;
    v2f bv; bv.x = Pp[(k0 * 4 + j) * WW + wpos];
            bv.y = Pp[((k0 + 1) * 4 + j) * WW + wpos];
    v8f cz = {};
    v8f d = __builtin_amdgcn_wmma_f32_16x16x4_f32(false, av, false, bv,
                                                  (short)0, cz, false, false);
    const int cwwA = cA * WW + wpos;
    const int cwwB = cwwA + WW;
    const size_t baseA = ((size_t)b * CWW + cwwA) * HH + j;
    const size_t baseB = ((size_t)b * CWW + cwwB) * HH + j;
    float psA = 0.f, psB = 0.f;
#pragma unroll
    for (int v = 0; v < 4; ++v) {
      float mA = mu[baseA + v * 4], sA = sg[baseA + v * 4];
      float dA = d[v] - mA;
      psA += expf(-dA * dA / (2.f * sA) - 0.5f * logf(sA) - HALF_LN2PI);
      float mB = mu[baseB + v * 4], sB = sg[baseB + v * 4];
      float dB = d[v + 4] - mB;
      psB += expf(-dB * dB / (2.f * sB) - 0.5f * logf(sB) - HALF_LN2PI);
    }
    psA += __shfl_xor(psA, 1, 4);  psA += __shfl_xor(psA, 2, 4);
    psB += __shfl_xor(psB, 1, 4);  psB += __shfl_xor(psB, 2, 4);
    if (j == 0) {
      float apA = aVb[cwwA] * psA;
      float apB = aVb[cwwB] * psB;
      Rrow[cwwA] = apA;
      Rrow[cwwB] = apB;
      den_loc += apA + apB;
    }
  }

  // make the global R stores visible before the normalize pass
  asm volatile("s_wait_storecnt 0" ::: "memory");
  red[threadIdx.x] = den_loc;
  __syncthreads();
  for (int s = 128; s > 0; s >>= 1) {
    if (threadIdx.x < s) red[threadIdx.x] += red[threadIdx.x + s];
    __syncthreads();
  }
  const float inv = 1.f / (red[0] + EPSF);
  for (int i = threadIdx.x; i < CWW; i += 256)
    Rrow[i] = Rrow[i] * inv + EPSF;
}

// ---------------------------------------------------------------------------
extern "C" void kernel_launch(void* const* d_in, const int* in_sizes, int n_in,
                              void* d_out, int out_size, void* d_ws, size_t ws_size,
                              hipStream_t stream) {
  const float* lam   = (const float*)d_in[0];
  const float* poses = (const float*)d_in[1];
  const float* act   = (const float*)d_in[2];
  const float* W     = (const float*)d_in[3];
  const float* bvv   = (const float*)d_in[4];
  const float* baa   = (const float*)d_in[5];
  float* out = (float*)d_out;
  float* ws  = (float*)d_ws;

  float* P  = ws;          // NP
  float* AP = P  + NP;     // NAP
  float* R  = AP + NAP;    // NR
  float* MU = R  + NR;     // NMU
  float* SG = MU + NMU;    // NMU
  float* AV = SG + NMU;    // NAV   (total ~35 MB)

  prep_kernel<<<(NR + 255) / 256, 256, 0, stream>>>(poses, act, P, AP, R);

  for (int it = 0; it < NITER; ++it) {
    float* muO = (it == NITER - 1) ? out : MU;
    float* aO  = (it == NITER - 1) ? (out + (size_t)NB * CWW * HH) : AV;
    stats_kernel<<<NB * 9, 256, 0, stream>>>(W, P, AP, R, bvv, baa, lam, muO, SG, aO);
    if (it != NITER - 1)
      rupdate_kernel<<<NB * BKK, 256, 0, stream>>>(W, P, MU, SG, AV, R);
  }
}